// DualChannelMPNN_72267119723176
// MI455X (gfx1250) — compile-verified
//
#include <hip/hip_runtime.h>
#include <math.h>

typedef __attribute__((ext_vector_type(2))) float v2f;
typedef __attribute__((ext_vector_type(8))) float v8f;

#define N_NODES 100000
#define N_EDGES 1600000
#define HDIM 64
#define G_GRAPHS 2048
#define DDIM 192      // 3H
#define T_MP 3
#define T_S2S 3
#define R_OUT 1024

// ---------------------------------------------------------------- WMMA helpers

__device__ __forceinline__ v8f wmma4(v2f a, v2f b, v8f c) {
  // D = A(16x4 f32) * B(4x16 f32) + C(16x16 f32)
  return __builtin_amdgcn_wmma_f32_16x16x4_f32(false, a, false, b, (short)0, c,
                                               false, false);
}

__device__ __forceinline__ v8f zero8() {
  v8f z = {0.f, 0.f, 0.f, 0.f, 0.f, 0.f, 0.f, 0.f};
  return z;
}

__device__ __forceinline__ float sigmoidf_(float x) {
  return 1.f / (1.f + expf(-x));
}

// A-fragment: row-major A, lane m=lane&15 holds rows; VGPR pair = K 2h,2h+1.
// Per-lane pointer pa = &A[(row0+m)*lda + 2h]; advance by 4 per k-step.
// B-fragment: lane n=lane&15, rows k+2h / k+2h+1 at column col0+n.

// ---------------------------------------------------------------- utility kernels

__global__ void fill_f32(float* p, float v, long n) {
  long i = (long)blockIdx.x * blockDim.x + threadIdx.x;
  if (i < n) p[i] = v;
}

__global__ void fill_i32(int* p, int v, long n) {
  long i = (long)blockIdx.x * blockDim.x + threadIdx.x;
  if (i < n) p[i] = v;
}

__global__ void copy2_kernel(const float* __restrict__ src, float* __restrict__ d0,
                             float* __restrict__ d1, long n) {
  long i = (long)blockIdx.x * blockDim.x + threadIdx.x;
  if (i < n) {
    float v = src[i];
    d0[i] = v;
    d1[i] = v;
  }
}

// ---------------------------------------------------------------- generic GEMM
// C[M,Ncols] = act( (A (+A2)) @ B + bias ) ; A [M,K], B [K,Ncols] row-major
// ACT: 0=none 1=relu 2=prelu(prelu[0])
template <bool HAS_A2, int ACT>
__global__ void gemm_kernel(const float* __restrict__ A, const float* __restrict__ A2,
                            const float* __restrict__ B, const float* __restrict__ bias,
                            float* __restrict__ C, int M, int K, int Ncols,
                            const float* __restrict__ prelu) {
  int lane = threadIdx.x & 31;
  int tile = blockIdx.x * (blockDim.x >> 5) + (threadIdx.x >> 5);
  int ntn = Ncols >> 4;
  int total = (M >> 4) * ntn;
  if (tile >= total) return;
  int mt = tile / ntn;
  int nt = tile - mt * ntn;
  long row0 = (long)mt << 4;
  int col0 = nt << 4;

  int n = lane & 15;
  int h = lane >> 4;

  const float* pa = A + (row0 + n) * (long)K + 2 * h;   // A rows indexed by lane&15
  const float* pa2 = HAS_A2 ? (A2 + (row0 + n) * (long)K + 2 * h) : nullptr;
  const float* pb0 = B + (long)(2 * h) * Ncols + col0 + n;
  const float* pb1 = pb0 + Ncols;
  const long bstep = 4 * (long)Ncols;

  v8f acc = zero8();
#pragma unroll 4
  for (int k = 0; k < K; k += 4) {
    v2f a = *(const v2f*)pa;
    if (HAS_A2) {
      v2f a2 = *(const v2f*)pa2;
      a.x += a2.x;
      a.y += a2.y;
      pa2 += 4;
    }
    v2f b;
    b.x = *pb0;
    b.y = *pb1;
    pa += 4;
    pb0 += bstep;
    pb1 += bstep;
    acc = wmma4(a, b, acc);
  }

  float bc = bias[col0 + n];
  float pav = (ACT == 2) ? prelu[0] : 0.f;
  float* pc = C + (row0 + h * 8) * (long)Ncols + col0 + n;
#pragma unroll
  for (int j = 0; j < 8; j++) {
    float v = acc[j] + bc;
    if (ACT == 1)
      v = fmaxf(v, 0.f);
    else if (ACT == 2)
      v = (v >= 0.f) ? v : pav * v;
    pc[(long)j * Ncols] = v;
  }
}

// ---------------------------------------------------------------- edge messages
// per-edge wave: msg = relu(hs[src] + (ef @ W_edge + b_edge)) ; aggr[dst] += msg
__global__ void edge_msg_kernel(const float* __restrict__ ef, const int* __restrict__ ei,
                                const float* __restrict__ hs, const float* __restrict__ We,
                                const float* __restrict__ be, float* __restrict__ aggr,
                                int Ecnt) {
  int lane = threadIdx.x & 31;
  long e = (long)blockIdx.x * (blockDim.x >> 5) + (threadIdx.x >> 5);
  if (e >= Ecnt) return;
  int src = ei[e];
  int dst = ei[(long)Ecnt + e];
  float efv = (lane < 16) ? ef[e * 16 + lane] : 0.f;
  float a0 = be[lane];
  float a1 = be[lane + 32];
#pragma unroll
  for (int k = 0; k < 16; k++) {
    float ek = __shfl(efv, k, 32);
    a0 = fmaf(ek, We[k * 64 + lane], a0);
    a1 = fmaf(ek, We[k * 64 + lane + 32], a1);
  }
  const float* hsr = hs + (long)src * 64;
  float m0 = fmaxf(hsr[lane] + a0, 0.f);
  float m1 = fmaxf(hsr[lane + 32] + a1, 0.f);
  atomicAdd(&aggr[(long)dst * 64 + lane], m0);
  atomicAdd(&aggr[(long)dst * 64 + lane + 32], m1);
}

// ---------------------------------------------------------------- per-node gates
__global__ void gates_kernel(const float* __restrict__ hs, const float* __restrict__ hr,
                             const float* __restrict__ am, const float* __restrict__ Wgs,
                             const float* __restrict__ bgs, const float* __restrict__ Wgr,
                             const float* __restrict__ bgr, float* __restrict__ xs,
                             float* __restrict__ xr, float* __restrict__ out_gate,
                             int Nn) {
  int lane = threadIdx.x & 31;
  long n = (long)blockIdx.x * (blockDim.x >> 5) + (threadIdx.x >> 5);
  if (n >= Nn) return;
  const float* hsr = hs + n * 64;
  const float* hrr = hr + n * 64;
  float vs = hsr[lane] * Wgs[lane] + hsr[lane + 32] * Wgs[lane + 32];
  float vr = hrr[lane] * Wgr[lane] + hrr[lane + 32] * Wgr[lane + 32];
#pragma unroll
  for (int off = 16; off; off >>= 1) {
    vs += __shfl_xor(vs, off, 32);
    vr += __shfl_xor(vr, off, 32);
  }
  float gs = sigmoidf_(vs + bgs[0]);
  float gr = sigmoidf_(vr + bgr[0]);
  float a0 = am[n * 64 + lane];
  float a1 = am[n * 64 + lane + 32];
  xs[n * 64 + lane] = gs * a0;
  xs[n * 64 + lane + 32] = gs * a1;
  xr[n * 64 + lane] = gr * a0;
  xr[n * 64 + lane + 32] = gr * a1;
  if (lane == 0) out_gate[n] = gr;
}

// ---------------------------------------------------------------- fused GRU cell
// Hn = GRU(X, Hp) with Wih/Whh [64 x 192], gate order r,z,n
__global__ void gru_kernel(const float* __restrict__ X, const float* __restrict__ Hp,
                           const float* __restrict__ Wih, const float* __restrict__ Whh,
                           const float* __restrict__ bih, const float* __restrict__ bhh,
                           float* __restrict__ Hn, int M) {
  int lane = threadIdx.x & 31;
  int tile = blockIdx.x * (blockDim.x >> 5) + (threadIdx.x >> 5);
  int total = (M >> 4) * 4;  // 64 output cols -> 4 col tiles
  if (tile >= total) return;
  int mt = tile >> 2;
  int cb = tile & 3;
  long row0 = (long)mt << 4;
  int col0 = cb << 4;

  int n = lane & 15;
  int h = lane >> 4;

  const float* px = X + (row0 + n) * 64 + 2 * h;
  const float* ph = Hp + (row0 + n) * 64 + 2 * h;
  const float* wi0 = Wih + (long)(2 * h) * 192 + col0 + n;  // row 2h
  const float* wi1 = wi0 + 192;                              // row 2h+1
  const float* wh0 = Whh + (long)(2 * h) * 192 + col0 + n;
  const float* wh1 = wh0 + 192;

  v8f aR = zero8(), aZ = zero8(), aI = zero8(), aH = zero8();
#pragma unroll 4
  for (int k = 0; k < 64; k += 4) {
    v2f ax = *(const v2f*)px;
    v2f ah = *(const v2f*)ph;
    v2f bri = {wi0[0], wi1[0]};
    v2f bzi = {wi0[64], wi1[64]};
    v2f bni = {wi0[128], wi1[128]};
    v2f brh = {wh0[0], wh1[0]};
    v2f bzh = {wh0[64], wh1[64]};
    v2f bnh = {wh0[128], wh1[128]};
    px += 4;
    ph += 4;
    wi0 += 4 * 192;
    wi1 += 4 * 192;
    wh0 += 4 * 192;
    wh1 += 4 * 192;
    aR = wmma4(ax, bri, aR);
    aR = wmma4(ah, brh, aR);
    aZ = wmma4(ax, bzi, aZ);
    aZ = wmma4(ah, bzh, aZ);
    aI = wmma4(ax, bni, aI);
    aH = wmma4(ah, bnh, aH);
  }

  int col = col0 + n;
  float brr = bih[col] + bhh[col];
  float bzz = bih[64 + col] + bhh[64 + col];
  float bin = bih[128 + col];
  float bhn = bhh[128 + col];
  const float* phc = Hp + (row0 + h * 8) * 64 + col;
  float* pout = Hn + (row0 + h * 8) * 64 + col;
#pragma unroll
  for (int j = 0; j < 8; j++) {
    float hold = phc[(long)j * 64];
    float r = sigmoidf_(aR[j] + brr);
    float z = sigmoidf_(aZ[j] + bzz);
    float nn = tanhf(aI[j] + bin + r * (aH[j] + bhn));
    pout[(long)j * 64] = (1.f - z) * nn + z * hold;
  }
}

// ---------------------------------------------------------------- fused LSTM cell
// gates = Q[G,384]@Wih[384,768] + Hp[G,192]@Whh[192,768] + biases; order i,f,g,o
__global__ void lstm_kernel(const float* __restrict__ Q, const float* __restrict__ Hp,
                            const float* __restrict__ Cp, const float* __restrict__ Wih,
                            const float* __restrict__ Whh, const float* __restrict__ bih,
                            const float* __restrict__ bhh, float* __restrict__ Hn,
                            float* __restrict__ Cn, int Gn) {
  int lane = threadIdx.x & 31;
  int tile = blockIdx.x * (blockDim.x >> 5) + (threadIdx.x >> 5);
  int total = (Gn >> 4) * 12;  // 192 output cols -> 12 col tiles
  if (tile >= total) return;
  int mt = tile / 12;
  int cb = tile - mt * 12;
  long row0 = (long)mt << 4;
  int col0 = cb << 4;

  int n = lane & 15;
  int h = lane >> 4;

  v8f ai = zero8(), af = zero8(), ag = zero8(), ao = zero8();

  {
    const float* pq = Q + (row0 + n) * 384 + 2 * h;
    const float* w0 = Wih + (long)(2 * h) * 768 + col0 + n;
    const float* w1 = w0 + 768;
#pragma unroll 4
    for (int k = 0; k < 384; k += 4) {
      v2f a = *(const v2f*)pq;
      v2f bi_ = {w0[0], w1[0]};
      v2f bf_ = {w0[192], w1[192]};
      v2f bg_ = {w0[384], w1[384]};
      v2f bo_ = {w0[576], w1[576]};
      pq += 4;
      w0 += 4 * 768;
      w1 += 4 * 768;
      ai = wmma4(a, bi_, ai);
      af = wmma4(a, bf_, af);
      ag = wmma4(a, bg_, ag);
      ao = wmma4(a, bo_, ao);
    }
  }
  {
    const float* ph = Hp + (row0 + n) * 192 + 2 * h;
    const float* w0 = Whh + (long)(2 * h) * 768 + col0 + n;
    const float* w1 = w0 + 768;
#pragma unroll 4
    for (int k = 0; k < 192; k += 4) {
      v2f a = *(const v2f*)ph;
      v2f bi_ = {w0[0], w1[0]};
      v2f bf_ = {w0[192], w1[192]};
      v2f bg_ = {w0[384], w1[384]};
      v2f bo_ = {w0[576], w1[576]};
      ph += 4;
      w0 += 4 * 768;
      w1 += 4 * 768;
      ai = wmma4(a, bi_, ai);
      af = wmma4(a, bf_, af);
      ag = wmma4(a, bg_, ag);
      ao = wmma4(a, bo_, ao);
    }
  }

  int col = col0 + n;
  float bi = bih[col] + bhh[col];
  float bf = bih[192 + col] + bhh[192 + col];
  float bg = bih[384 + col] + bhh[384 + col];
  float bo = bih[576 + col] + bhh[576 + col];
  const float* pcold = Cp + (row0 + h * 8) * 192 + col;
  float* pcn = Cn + (row0 + h * 8) * 192 + col;
  float* phn = Hn + (row0 + h * 8) * 192 + col;
#pragma unroll
  for (int j = 0; j < 8; j++) {
    float cold = pcold[(long)j * 192];
    float iv = sigmoidf_(ai[j] + bi);
    float fv = sigmoidf_(af[j] + bf);
    float gv = tanhf(ag[j] + bg);
    float ov = sigmoidf_(ao[j] + bo);
    float cn = fv * cold + iv * gv;
    pcn[(long)j * 192] = cn;
    phn[(long)j * 192] = ov * tanhf(cn);
  }
}

// ---------------------------------------------------------------- readout pieces

__global__ void concat_kernel(const float* __restrict__ x0, const float* __restrict__ hs,
                              const float* __restrict__ hr, float* __restrict__ xro,
                              long total) {
  long i = (long)blockIdx.x * blockDim.x + threadIdx.x;
  if (i >= total) return;
  long nd = i / 192;
  int c = (int)(i - nd * 192);
  float v = (c < 64) ? x0[nd * 64 + c]
                     : ((c < 128) ? hs[nd * 64 + (c - 64)] : hr[nd * 64 + (c - 128)]);
  xro[i] = v;
}

__global__ void hist_kernel(const int* __restrict__ batch, int* __restrict__ counts,
                            int Nn) {
  long i = (long)blockIdx.x * blockDim.x + threadIdx.x;
  if (i < Nn) atomicAdd(&counts[batch[i]], 1);
}

__global__ void scan_kernel(const int* __restrict__ counts, int* __restrict__ starts,
                            int Gn) {
  if (blockIdx.x == 0 && threadIdx.x == 0) {
    int s = 0;
    for (int g = 0; g < Gn; g++) {
      starts[g] = s;
      s += counts[g];
    }
    starts[Gn] = s;
  }
}

// e[n] = dot(x_ro[n], q[batch[n]]) over 192 dims (one wave per node)
__global__ void edot_kernel(const float* __restrict__ xro, const float* __restrict__ q,
                            const int* __restrict__ batch, float* __restrict__ e_out,
                            int Nn) {
  int lane = threadIdx.x & 31;
  long nd = (long)blockIdx.x * (blockDim.x >> 5) + (threadIdx.x >> 5);
  if (nd >= Nn) return;
  int b = batch[nd];
  const float* xr = xro + nd * 192;
  const float* qq = q + (long)b * 192;
  float v = 0.f;
#pragma unroll
  for (int c = 0; c < 6; c++) v = fmaf(xr[lane + 32 * c], qq[lane + 32 * c], v);
#pragma unroll
  for (int off = 16; off; off >>= 1) v += __shfl_xor(v, off, 32);
  if (lane == 0) e_out[nd] = v;
}

// one workgroup per graph: softmax over its (sorted, contiguous) nodes, then
// r[g,c] = sum_n a[n] * x_ro[n,c]
__global__ void graph_softmax_kernel(float* __restrict__ ebuf,
                                     const float* __restrict__ xro,
                                     const int* __restrict__ starts,
                                     float* __restrict__ r) {
  __shared__ float red[256];
  __shared__ float m_sh, d_sh;
  int g = blockIdx.x;
  int t = threadIdx.x;
  int ns = starts[g], ne = starts[g + 1];

  float m = -INFINITY;
  for (int nn = ns + t; nn < ne; nn += 256) m = fmaxf(m, ebuf[nn]);
  red[t] = m;
  __syncthreads();
  for (int s = 128; s > 0; s >>= 1) {
    if (t < s) red[t] = fmaxf(red[t], red[t + s]);
    __syncthreads();
  }
  if (t == 0) m_sh = red[0];
  __syncthreads();
  m = m_sh;

  float sum = 0.f;
  for (int nn = ns + t; nn < ne; nn += 256) {
    float ex = expf(ebuf[nn] - m);
    ebuf[nn] = ex;
    sum += ex;
  }
  red[t] = sum;
  __syncthreads();
  for (int s = 128; s > 0; s >>= 1) {
    if (t < s) red[t] += red[t + s];
    __syncthreads();
  }
  if (t == 0) d_sh = red[0];
  __syncthreads();
  float inv = (d_sh > 0.f) ? 1.f / d_sh : 0.f;

  for (int c = t; c < 192; c += 256) {
    float acc = 0.f;
    for (int nn = ns; nn < ne; nn++) acc = fmaf(ebuf[nn], xro[(long)nn * 192 + c], acc);
    r[(long)g * 192 + c] = acc * inv;
  }
}

__global__ void qstar_kernel(const float* __restrict__ h, const float* __restrict__ r,
                             float* __restrict__ q, long total) {
  long i = (long)blockIdx.x * blockDim.x + threadIdx.x;
  if (i >= total) return;
  long g = i / 384;
  int c = (int)(i - g * 384);
  q[i] = (c < 192) ? h[g * 192 + c] : r[g * 192 + (c - 192)];
}

// ---------------------------------------------------------------- host launcher

static inline int blocksElem(long n, int bs) { return (int)((n + bs - 1) / bs); }
static inline int blocksWave(long waves) { return (int)((waves + 7) / 8); }  // 256 thr = 8 waves

extern "C" void kernel_launch(void* const* d_in, const int* in_sizes, int n_in,
                              void* d_out, int out_size, void* d_ws, size_t ws_size,
                              hipStream_t stream) {
  const float* node_feats = (const float*)d_in[0];
  const float* edge_feats = (const float*)d_in[1];
  const int* edge_index = (const int*)d_in[2];
  const int* batch = (const int*)d_in[3];
  const float* W_proj = (const float*)d_in[4];
  const float* b_proj = (const float*)d_in[5];
  const float* W_edge = (const float*)d_in[6];
  const float* b_edge = (const float*)d_in[7];
  const float* W_gine = (const float*)d_in[8];
  const float* b_gine = (const float*)d_in[9];
  const float* Wg_s = (const float*)d_in[10];
  const float* bg_s = (const float*)d_in[11];
  const float* Wg_r = (const float*)d_in[12];
  const float* bg_r = (const float*)d_in[13];
  const float* Ws_ih = (const float*)d_in[14];
  const float* Ws_hh = (const float*)d_in[15];
  const float* bs_ih = (const float*)d_in[16];
  const float* bs_hh = (const float*)d_in[17];
  const float* Wr_ih = (const float*)d_in[18];
  const float* Wr_hh = (const float*)d_in[19];
  const float* br_ih = (const float*)d_in[20];
  const float* br_hh = (const float*)d_in[21];
  const float* Wl_ih = (const float*)d_in[22];
  const float* Wl_hh = (const float*)d_in[23];
  const float* bl_ih = (const float*)d_in[24];
  const float* bl_hh = (const float*)d_in[25];
  const float* W_sp = (const float*)d_in[26];
  const float* b_sp = (const float*)d_in[27];
  const float* a_pr = (const float*)d_in[28];

  float* out = (float*)d_out;
  float* out_gates = out + (long)G_GRAPHS * R_OUT;  // [T_MP, N, 1]

  // ---- workspace carve-up
  float* w = (float*)d_ws;
  const size_t NF = (size_t)N_NODES * HDIM;
  float* x0 = w;        w += NF;
  float* hsA = w;       w += NF;
  float* hsB = w;       w += NF;
  float* hrA = w;       w += NF;
  float* hrB = w;       w += NF;
  float* aggr = w;      w += NF;
  float* am = w;        w += NF;
  float* xs = w;        w += NF;
  float* xr = w;        w += NF;
  float* xro = w;       w += (size_t)N_NODES * DDIM;
  float* ebuf = w;      w += N_NODES;
  float* qstar = w;     w += (size_t)G_GRAPHS * 2 * DDIM;
  float* hLA = w;       w += (size_t)G_GRAPHS * DDIM;
  float* hLB = w;       w += (size_t)G_GRAPHS * DDIM;
  float* cLA = w;       w += (size_t)G_GRAPHS * DDIM;
  float* cLB = w;       w += (size_t)G_GRAPHS * DDIM;
  float* rbuf = w;      w += (size_t)G_GRAPHS * DDIM;
  int* counts = (int*)w;
  int* starts = counts + G_GRAPHS;  // G+1 entries

  // ---- 1. project node feats: x0 = relu(node_feats @ W_proj + b_proj)
  {
    long tiles = (long)(N_NODES / 16) * (HDIM / 16);
    gemm_kernel<false, 1><<<blocksWave(tiles), 256, 0, stream>>>(
        node_feats, nullptr, W_proj, b_proj, x0, N_NODES, 128, HDIM, nullptr);
  }
  copy2_kernel<<<blocksElem(NF, 256), 256, 0, stream>>>(x0, hsA, hrA, (long)NF);

  float* hs_cur = hsA;
  float* hs_nxt = hsB;
  float* hr_cur = hrA;
  float* hr_nxt = hrB;

  // ---- 2. message-passing steps
  for (int t = 0; t < T_MP; t++) {
    fill_f32<<<blocksElem(NF, 256), 256, 0, stream>>>(aggr, 0.f, (long)NF);
    edge_msg_kernel<<<blocksWave(N_EDGES), 256, 0, stream>>>(
        edge_feats, edge_index, hs_cur, W_edge, b_edge, aggr, N_EDGES);
    {
      long tiles = (long)(N_NODES / 16) * (HDIM / 16);
      gemm_kernel<true, 1><<<blocksWave(tiles), 256, 0, stream>>>(
          hs_cur, aggr, W_gine, b_gine, am, N_NODES, HDIM, HDIM, nullptr);
    }
    gates_kernel<<<blocksWave(N_NODES), 256, 0, stream>>>(
        hs_cur, hr_cur, am, Wg_s, bg_s, Wg_r, bg_r, xs, xr,
        out_gates + (long)t * N_NODES, N_NODES);
    {
      long tiles = (long)(N_NODES / 16) * 4;
      gru_kernel<<<blocksWave(tiles), 256, 0, stream>>>(xs, hs_cur, Ws_ih, Ws_hh,
                                                        bs_ih, bs_hh, hs_nxt, N_NODES);
      gru_kernel<<<blocksWave(tiles), 256, 0, stream>>>(xr, hr_cur, Wr_ih, Wr_hh,
                                                        br_ih, br_hh, hr_nxt, N_NODES);
    }
    float* tmp;
    tmp = hs_cur; hs_cur = hs_nxt; hs_nxt = tmp;
    tmp = hr_cur; hr_cur = hr_nxt; hr_nxt = tmp;
  }

  // ---- 3. x_ro = concat(x0, hs, hr)
  concat_kernel<<<blocksElem((long)N_NODES * DDIM, 256), 256, 0, stream>>>(
      x0, hs_cur, hr_cur, xro, (long)N_NODES * DDIM);

  // ---- 4. graph segment offsets (batch is sorted)
  fill_i32<<<blocksElem(G_GRAPHS, 256), 256, 0, stream>>>(counts, 0, G_GRAPHS);
  hist_kernel<<<blocksElem(N_NODES, 256), 256, 0, stream>>>(batch, counts, N_NODES);
  scan_kernel<<<1, 1, 0, stream>>>(counts, starts, G_GRAPHS);

  // ---- 5. Set2Set
  fill_f32<<<blocksElem((long)G_GRAPHS * 2 * DDIM, 256), 256, 0, stream>>>(
      qstar, 0.f, (long)G_GRAPHS * 2 * DDIM);
  fill_f32<<<blocksElem((long)G_GRAPHS * DDIM, 256), 256, 0, stream>>>(
      hLA, 0.f, (long)G_GRAPHS * DDIM);
  fill_f32<<<blocksElem((long)G_GRAPHS * DDIM, 256), 256, 0, stream>>>(
      cLA, 0.f, (long)G_GRAPHS * DDIM);

  float* hL_cur = hLA;
  float* hL_nxt = hLB;
  float* cL_cur = cLA;
  float* cL_nxt = cLB;

  for (int s = 0; s < T_S2S; s++) {
    long tiles = (long)(G_GRAPHS / 16) * (DDIM / 16);
    lstm_kernel<<<blocksWave(tiles), 256, 0, stream>>>(
        qstar, hL_cur, cL_cur, Wl_ih, Wl_hh, bl_ih, bl_hh, hL_nxt, cL_nxt, G_GRAPHS);
    float* tmp;
    tmp = hL_cur; hL_cur = hL_nxt; hL_nxt = tmp;
    tmp = cL_cur; cL_cur = cL_nxt; cL_nxt = tmp;

    edot_kernel<<<blocksWave(N_NODES), 256, 0, stream>>>(xro, hL_cur, batch, ebuf,
                                                         N_NODES);
    graph_softmax_kernel<<<G_GRAPHS, 256, 0, stream>>>(ebuf, xro, starts, rbuf);
    qstar_kernel<<<blocksElem((long)G_GRAPHS * 2 * DDIM, 256), 256, 0, stream>>>(
        hL_cur, rbuf, qstar, (long)G_GRAPHS * 2 * DDIM);
  }

  // ---- 6. final projection + PReLU
  {
    long tiles = (long)(G_GRAPHS / 16) * (R_OUT / 16);
    gemm_kernel<false, 2><<<blocksWave(tiles), 256, 0, stream>>>(
        qstar, nullptr, W_sp, b_sp, out, G_GRAPHS, 2 * DDIM, a_pr ? R_OUT : R_OUT, a_pr);
  }
}